// ModernBertCanonLayer_4020089389282
// MI455X (gfx1250) — compile-verified
//
#include <hip/hip_runtime.h>
#include <hip/hip_bf16.h>

#define T_TOT   65536
#define CH      768
#define KS      5
#define RAD     2
#define NSEQP1  17      // cu_seqlens length (NSEQ=16 + 1)
#define TOK_TILE 16
#define EXT     (TOK_TILE + 2 * RAD)   // 20 rows incl. halo
#define C4      (CH / 4)               // 192 float4 per row

__global__ __launch_bounds__(C4) void canon_varlen_dwconv_kernel(
    const float* __restrict__ x,      // [T, C]
    const int*   __restrict__ cu,     // [17]
    const float* __restrict__ w,      // [C, K]
    const float* __restrict__ bias,   // [C]
    float*       __restrict__ out)    // [T, C]
{
    __shared__ float sx[EXT * CH];    // 20*768*4 B = 61440 B
    __shared__ int   ssid[EXT];

    const int tid = threadIdx.x;                 // 0..191 = float4 channel group
    const int t0  = blockIdx.x * TOK_TILE;

    // ---- stage 20 rows global -> LDS with CDNA5 async copies (ASYNCcnt) ----
    #pragma unroll
    for (int r = 0; r < EXT; ++r) {
        int tok   = t0 + r - RAD;
        int tokCl = tok < 0 ? 0 : (tok >= T_TOT ? T_TOT - 1 : tok);
        const float* gsrc = x + (size_t)tokCl * CH + (size_t)tid * 4;
        unsigned long long gaddr = (unsigned long long)(size_t)gsrc;
        unsigned ldst = (unsigned)(size_t)(&sx[r * CH + tid * 4]); // LDS byte offset
        asm volatile("global_load_async_to_lds_b128 %0, %1, off"
                     :: "v"(ldst), "v"(gaddr)
                     : "memory");
    }

    // ---- sequence ids for the extended token range (sentinel -1 if OOB) ----
    if (tid < EXT) {
        int tok = t0 + tid - RAD;
        int sid = -1;
        if (tok >= 0 && tok < T_TOT) {
            sid = 0;
            #pragma unroll
            for (int j = 1; j < NSEQP1; ++j)
                sid += (cu[j] <= tok) ? 1 : 0;   // searchsorted(cu[1:], tok, 'right')
        }
        ssid[tid] = sid;
    }

    // ---- per-thread taps + bias for its 4 channels ----
    const int c0 = tid * 4;
    float wreg[KS][4];
    float breg[4];
    #pragma unroll
    for (int j = 0; j < 4; ++j) {
        breg[j] = bias[c0 + j];
        #pragma unroll
        for (int k = 0; k < KS; ++k)
            wreg[k][j] = w[(c0 + j) * KS + k];
    }

    // drain async copies, then make LDS visible to the whole workgroup
    asm volatile("s_wait_asynccnt 0" ::: "memory");
    __syncthreads();

    // ---- rolling 5-row register window over the LDS tile ----
    const float4* sx4 = (const float4*)sx;
    float4 win[KS];
    int    sidw[KS];
    #pragma unroll
    for (int k = 0; k < KS - 1; ++k) {
        win[k]  = sx4[k * C4 + tid];
        sidw[k] = ssid[k];
    }

    float4* out4 = (float4*)out;
    #pragma unroll
    for (int tok = 0; tok < TOK_TILE; ++tok) {
        win[KS - 1]  = sx4[(tok + KS - 1) * C4 + tid];
        sidw[KS - 1] = ssid[tok + KS - 1];

        const int    mys = sidw[RAD];
        const float4 xc  = win[RAD];

        float o0 = xc.x + breg[0];
        float o1 = xc.y + breg[1];
        float o2 = xc.z + breg[2];
        float o3 = xc.w + breg[3];

        #pragma unroll
        for (int k = 0; k < KS; ++k) {
            const bool v = (sidw[k] == mys);
            o0 = fmaf(v ? win[k].x : 0.0f, wreg[k][0], o0);
            o1 = fmaf(v ? win[k].y : 0.0f, wreg[k][1], o1);
            o2 = fmaf(v ? win[k].z : 0.0f, wreg[k][2], o2);
            o3 = fmaf(v ? win[k].w : 0.0f, wreg[k][3], o3);
        }

        float4 res; res.x = o0; res.y = o1; res.z = o2; res.w = o3;
        out4[(size_t)(t0 + tok) * C4 + tid] = res;

        // shift window
        #pragma unroll
        for (int k = 0; k < KS - 1; ++k) {
            win[k]  = win[k + 1];
            sidw[k] = sidw[k + 1];
        }
    }
}

extern "C" void kernel_launch(void* const* d_in, const int* in_sizes, int n_in,
                              void* d_out, int out_size, void* d_ws, size_t ws_size,
                              hipStream_t stream) {
    (void)in_sizes; (void)n_in; (void)out_size; (void)d_ws; (void)ws_size;
    const float* x    = (const float*)d_in[0];
    const int*   cu   = (const int*)  d_in[1];
    const float* w    = (const float*)d_in[2];
    const float* bias = (const float*)d_in[3];
    float*       out  = (float*)d_out;

    dim3 grid(T_TOT / TOK_TILE);   // 4096 blocks
    dim3 block(C4);                // 192 threads = 6 wave32
    canon_varlen_dwconv_kernel<<<grid, block, 0, stream>>>(x, cu, w, bias, out);
}